// TropicalDense_41592463294676
// MI455X (gfx1250) — compile-verified
//
#include <hip/hip_runtime.h>
#include <stdint.h>

// Problem sizes (fixed by the reference).
constexpr int BB = 1024;  // batch
constexpr int DD = 512;   // features
constexpr int OO = 256;   // outputs

// Tiling: 32x32 output tile per block, D streamed in chunks of 64.
constexpr int BT = 32;    // rows per block
constexpr int OT = 32;    // output cols per block (one per lane)
constexpr int DC = 64;    // d-chunk streamed through LDS
constexpr int NC = DD / DC;           // 8 chunks
constexpr int XCHUNK = BT * DC;       // 2048 floats (8 KB)
constexpr int KCHUNK = 2 * DC * OT;   // 4096 floats (16 KB)
constexpr int BUF = XCHUNK + KCHUNK;  // 6144 floats (24 KB) per buffer

// ---- CDNA5 async global->LDS (ASYNCcnt path), inline asm (toolchain-portable) ----
__device__ __forceinline__ void async_b128_to_lds(uint32_t lds_byte,
                                                  uint32_t goff_byte,
                                                  const void* sbase) {
  // saddr form: vdst = LDS byte address, vaddr = 32-bit byte offset, saddr = SGPR64 base
  asm volatile("global_load_async_to_lds_b128 %0, %1, %2"
               :: "v"(lds_byte), "v"(goff_byte), "s"(sbase) : "memory");
}

__device__ __forceinline__ void wait_async0() {
  asm volatile("s_wait_asynccnt 0" ::: "memory");
}

// Issue one d-chunk's loads (X tile + K tile) into LDS buffer `buf`.
__device__ __forceinline__ void issue_chunk(const float* __restrict__ Xg,
                                            const float* __restrict__ Kg,
                                            int b0, int o0, int c, int buf,
                                            int tid, uint32_t lds_base_byte) {
  // X tile: 2048 floats -> 2 x b128 per thread.
  // e = tid*4 + j*1024 ; r = e>>6 ; dd = e&63 ; LDS = buf*BUF + e ; G = (b0+r)*DD + c*DC + dd
#pragma unroll
  for (int j = 0; j < 2; ++j) {
    const int e  = tid * 4 + j * 1024;
    const int r  = e >> 6;
    const int dd = e & 63;
    const uint32_t g = (uint32_t)(((b0 + r) * DD + c * DC + dd) * 4);
    const uint32_t l = lds_base_byte + (uint32_t)((buf * BUF + e) * 4);
    async_b128_to_lds(l, g, Xg);
  }
  // K tile: 4096 floats -> 4 x b128 per thread.
  // e = s*2048 + dd*32 + o ; G = s*DD*OO + (c*DC+dd)*OO + o0 + o
#pragma unroll
  for (int j = 0; j < 4; ++j) {
    const int e   = tid * 4 + j * 1024;
    const int s   = e >> 11;
    const int rem = e & 2047;
    const int dd  = rem >> 5;
    const int o   = rem & 31;
    const uint32_t g = (uint32_t)((s * DD * OO + (c * DC + dd) * OO + o0 + o) * 4);
    const uint32_t l = lds_base_byte + (uint32_t)((buf * BUF + XCHUNK + e) * 4);
    async_b128_to_lds(l, g, Kg);
  }
}

__global__ __launch_bounds__(256) void tropical_dense_kernel(
    const float* __restrict__ X, const float* __restrict__ K,
    float* __restrict__ out) {
  __shared__ __align__(16) float lds[2][BUF];   // 48 KB, double buffered

  const int tid = threadIdx.x;
  const int b0  = blockIdx.x * BT;   // 32 row tiles
  const int o0  = blockIdx.y * OT;   // 8 col tiles
  const int ox  = tid & 31;          // lane -> output column
  const int r0  = (tid >> 5) * 4;    // wave -> 4 rows

  // LDS byte address of the shared array (low 32 bits of the shared-aperture
  // flat address are the LDS offset on gfx1250).
  const uint32_t lds_base_byte = (uint32_t)(uintptr_t)(&lds[0][0]);

  float acc[4];
#pragma unroll
  for (int r = 0; r < 4; ++r) acc[r] = -__builtin_huge_valf();

  // Prime the pipeline: chunk 0 into buffer 0.
  issue_chunk(X, K, b0, o0, /*c=*/0, /*buf=*/0, tid, lds_base_byte);
  wait_async0();
  __syncthreads();

  for (int c = 0; c < NC; ++c) {
    const int cur = c & 1;
    // Prefetch next chunk into the other buffer while we compute on this one.
    if (c + 1 < NC)
      issue_chunk(X, K, b0, o0, c + 1, (c + 1) & 1, tid, lds_base_byte);

    const float* __restrict__ Xs  = &lds[cur][0];
    const float* __restrict__ Kp0 = &lds[cur][XCHUNK];
    const float* __restrict__ Kp1 = &lds[cur][XCHUNK + DC * OT];

    for (int dd = 0; dd < DC; dd += 4) {
      float xv[4][4];
#pragma unroll
      for (int r = 0; r < 4; ++r) {
        const float4 v = *(const float4*)(Xs + (r0 + r) * DC + dd);  // LDS broadcast b128
        xv[r][0] = v.x; xv[r][1] = v.y; xv[r][2] = v.z; xv[r][3] = v.w;
      }
#pragma unroll
      for (int q = 0; q < 4; ++q) {
        const float k0 = Kp0[(dd + q) * OT + ox];
        const float k1 = Kp1[(dd + q) * OT + ox];
#pragma unroll
        for (int r = 0; r < 4; ++r) {
          const float x  = xv[r][q];
          const float t0 = x + k0;    // +X branch
          const float t1 = k1 - x;    // -X branch
          acc[r] = fmaxf(acc[r], fmaxf(t0, t1));  // v_max_num / v_max3_num_f32
        }
      }
    }

    // Drain this iteration's prefetch, then make the buffer swap safe.
    wait_async0();
    __syncthreads();
  }

  // Coalesced store of the 32x32 tile (4 rows per thread, lane = column).
#pragma unroll
  for (int r = 0; r < 4; ++r)
    out[(b0 + r0 + r) * OO + o0 + ox] = acc[r];
}

extern "C" void kernel_launch(void* const* d_in, const int* in_sizes, int n_in,
                              void* d_out, int out_size, void* d_ws, size_t ws_size,
                              hipStream_t stream) {
  const float* X = (const float*)d_in[0];     // [1024, 512] f32
  const float* K = (const float*)d_in[1];     // [2, 512, 256] f32
  float* out = (float*)d_out;                 // [1024, 256] f32
  (void)in_sizes; (void)n_in; (void)out_size; (void)d_ws; (void)ws_size;

  dim3 grid(BB / BT, OO / OT);  // (32, 8) = 256 workgroups
  dim3 block(256);              // 8 wave32s
  tropical_dense_kernel<<<grid, block, 0, stream>>>(X, K, out);
}